// Attention_84851373900348
// MI455X (gfx1250) — compile-verified
//
#include <hip/hip_runtime.h>
#include <math.h>

typedef __attribute__((ext_vector_type(2))) float v2f;
typedef __attribute__((ext_vector_type(4))) float v4f;
typedef __attribute__((ext_vector_type(8))) float v8f;

#define HEADS   8
#define DIM     64
#define EDGES   400000
#define HE      (HEADS * EDGES)         // 3,200,000 edges total
#define NSEG    (50000 * HEADS)         // 400,000 softmax segments

// Sign-split float atomic max using native integer atomics.
// Requires seg_max initialized to -inf (0xFF800000).
__device__ inline void atomicMaxF(float* addr, float v) {
    if (v >= 0.0f) atomicMax((int*)addr, __float_as_int(v));
    else           atomicMin((unsigned int*)addr, __float_as_uint(v));
}

__global__ void gat_init(float* __restrict__ seg_max, float* __restrict__ seg_sum) {
    int i = blockIdx.x * blockDim.x + threadIdx.x;
    if (i < NSEG) {
        ((int*)seg_max)[i] = 0xFF800000; // -inf
        seg_sum[i] = 0.0f;
    }
}

// Pass 1: e = sum_d xi*xj*w via V_WMMA_F32_16X16X4_F32 (full fp32 path).
// One wave32 handles a 16-edge tile. K-dims are permuted so each lane half
// owns a contiguous 32-float slice of its row: half 0 -> dims 0..31 (K-slots
// 0,1 of each chunk), half 1 -> dims 32..63 (K-slots 2,3). This lets every
// lane fetch with b128 loads; each float4 feeds two consecutive WMMAs.
// B = w broadcast to all 16 columns -> every output column equals e[rows].
__global__ __launch_bounds__(256) void gat_pass1(
        const float* __restrict__ x_i, const float* __restrict__ x_j,
        const float* __restrict__ a,   const long long* __restrict__ ei,
        float* __restrict__ ws_e,      float* __restrict__ seg_max) {
    __shared__ float lds_w[HEADS * DIM]; // w[h][d] = a_l[h][d]*a_r[h][d]
    __shared__ float lds_e[8 * 16];      // per-wave 16-edge result exchange

    const int tid = threadIdx.x;
    for (int t = tid; t < HEADS * DIM; t += 256) {
        int h = t >> 6, d = t & 63;
        lds_w[t] = a[h * 128 + d] * a[h * 128 + 64 + d];
    }
    __syncthreads();

    const int wid   = tid >> 5;
    const int lane  = tid & 31;
    const int m     = lane & 15;       // tile row (edge within tile)
    const int khalf = lane >> 4;       // 0: K-slots {0,1}, 1: K-slots {2,3}
    const int doff  = khalf * 32;      // this half's contiguous dim slice

    const long long base = ((long long)blockIdx.x * 8 + wid) * 16; // first edge of tile
    const int h = (int)(base / EDGES);         // EDGES % 16 == 0 -> tile head-uniform
    const float* wrow = &lds_w[h * DIM + doff];

    const size_t rowoff = (size_t)(base + m) * DIM + doff;
    const float* xi_p = x_i + rowoff;
    const float* xj_p = x_j + rowoff;

    v8f acc = {0.f, 0.f, 0.f, 0.f, 0.f, 0.f, 0.f, 0.f};
    #pragma unroll
    for (int t = 0; t < 8; ++t) {                       // 8 x float4 -> 16 WMMAs
        v4f xi = *(const v4f*)(xi_p + 4 * t);           // global_load_b128
        v4f xj = *(const v4f*)(xj_p + 4 * t);           // global_load_b128
        v4f w4 = *(const v4f*)(wrow + 4 * t);           // ds_load_b128
        v4f p4 = xi * xj;                               // 2x v_pk_mul_f32
        v2f pA0; pA0.x = p4.x; pA0.y = p4.y;
        v2f bB0; bB0.x = w4.x; bB0.y = w4.y;
        acc = __builtin_amdgcn_wmma_f32_16x16x4_f32(
                false, pA0, false, bB0, (short)0, acc, false, false);
        v2f pA1; pA1.x = p4.z; pA1.y = p4.w;
        v2f bB1; bB1.x = w4.z; bB1.y = w4.w;
        acc = __builtin_amdgcn_wmma_f32_16x16x4_f32(
                false, pA1, false, bB1, (short)0, acc, false, false);
    }

    // D layout: VGPR r holds M = r + 8*khalf (same value in every lane of the half).
    if (m == 0) {
        float* dst = &lds_e[wid * 16 + khalf * 8];
        #pragma unroll
        for (int r = 0; r < 8; ++r) dst[r] = acc[r];
    }
    __syncthreads();

    if (lane < 16) {
        const long long edge = base + lane;
        const float e = lds_e[wid * 16 + lane];
        ws_e[edge] = e;                                 // coalesced 64B per wave
        const int seg = (int)ei[HE + edge];             // edge_index row 1
        atomicMaxF(&seg_max[seg], e);                   // L2-resident, low contention
    }
}

// Pass 2: z = exp(e - seg_max[seg]); in-place into ws_e; seg_sum += z.
// 4 edges per thread: float4 for e/z, contiguous int64 index loads.
__global__ void gat_pass2(const long long* __restrict__ ei,
                          const float* __restrict__ seg_max,
                          float* __restrict__ ws_e,
                          float* __restrict__ seg_sum) {
    int i4 = (blockIdx.x * blockDim.x + threadIdx.x) * 4;
    if (i4 < HE) {                                      // HE % 4 == 0
        v4f e = *(const v4f*)(ws_e + i4);
        int seg[4];
        #pragma unroll
        for (int k = 0; k < 4; ++k) seg[k] = (int)ei[HE + i4 + k];
        v4f z;
        z.x = expf(e.x - seg_max[seg[0]]);
        z.y = expf(e.y - seg_max[seg[1]]);
        z.z = expf(e.z - seg_max[seg[2]]);
        z.w = expf(e.w - seg_max[seg[3]]);
        *(v4f*)(ws_e + i4) = z;
        atomicAdd(&seg_sum[seg[0]], z.x);               // native global_atomic_add_f32
        atomicAdd(&seg_sum[seg[1]], z.y);
        atomicAdd(&seg_sum[seg[2]], z.z);
        atomicAdd(&seg_sum[seg[3]], z.w);
    }
}

// Pass 3: alpha = z / (seg_sum[seg] + 1e-16).
__global__ void gat_pass3(const long long* __restrict__ ei,
                          const float* __restrict__ seg_sum,
                          const float* __restrict__ ws_e,
                          float* __restrict__ out) {
    int i4 = (blockIdx.x * blockDim.x + threadIdx.x) * 4;
    if (i4 < HE) {
        v4f z = *(const v4f*)(ws_e + i4);
        int seg[4];
        #pragma unroll
        for (int k = 0; k < 4; ++k) seg[k] = (int)ei[HE + i4 + k];
        v4f r;
        r.x = z.x / (seg_sum[seg[0]] + 1e-16f);
        r.y = z.y / (seg_sum[seg[1]] + 1e-16f);
        r.z = z.z / (seg_sum[seg[2]] + 1e-16f);
        r.w = z.w / (seg_sum[seg[3]] + 1e-16f);
        *(v4f*)(out + i4) = r;
    }
}

extern "C" void kernel_launch(void* const* d_in, const int* in_sizes, int n_in,
                              void* d_out, int out_size, void* d_ws, size_t ws_size,
                              hipStream_t stream) {
    const float*     x_i = (const float*)d_in[0];
    const float*     x_j = (const float*)d_in[1];
    const float*     a   = (const float*)d_in[2];
    const long long* ei  = (const long long*)d_in[3];
    float* out = (float*)d_out;

    float* ws_e    = (float*)d_ws;          // HE floats (e, then z in-place)
    float* seg_max = ws_e + HE;             // NSEG floats
    float* seg_sum = seg_max + NSEG;        // NSEG floats  (total 16 MB)

    (void)in_sizes; (void)n_in; (void)out_size; (void)ws_size;

    gat_init <<<(NSEG + 255) / 256, 256, 0, stream>>>(seg_max, seg_sum);
    gat_pass1<<<HE / (16 * 8), 256, 0, stream>>>(x_i, x_j, a, ei, ws_e, seg_max);
    gat_pass2<<<HE / (4 * 256), 256, 0, stream>>>(ei, seg_max, ws_e, seg_sum);
    gat_pass3<<<HE / (4 * 256), 256, 0, stream>>>(ei, seg_sum, ws_e, out);
}